// ExpMessageAggregator_63024350102031
// MI455X (gfx1250) — compile-verified
//
#include <hip/hip_runtime.h>
#include <hip/hip_bf16.h>
#include <stdint.h>
#include <utility>

#define D 256          // feature dim (fixed by reference)
#define CHUNK 32       // messages staged per async batch (1 LDS buffer = 32 KB)
#define SPAN 256       // contiguous messages per block (multiple of CHUNK)
#define ROWB (D * 4)   // bytes per message row = 1024

__device__ __forceinline__ void flush_acc(float* __restrict__ agg, int seg, int col,
                                          float4 a) {
  float* p = agg + (size_t)seg * D + col;
  atomicAdd(p + 0, a.x);
  atomicAdd(p + 1, a.y);
  atomicAdd(p + 2, a.z);
  atomicAdd(p + 3, a.w);
}

// One async row copy; INST_OFFSET is added to BOTH the LDS and global address
// (CDNA5 ISA), so a whole contiguous chunk needs only one
// (lds, global) base pair + compile-time immediate offsets.
template <int OFS>
__device__ __forceinline__ void issue_row(unsigned la, unsigned long long ga) {
  asm volatile("global_load_async_to_lds_b128 %0, %1, off offset:%2 th:TH_LOAD_NT"
               :: "v"(la), "v"(ga), "n"(OFS) : "memory");
}

template <int... I>
__device__ __forceinline__ void issue_chunk_impl(unsigned la, unsigned long long ga,
                                                 std::integer_sequence<int, I...>) {
  (issue_row<I * ROWB>(la, ga), ...);
}

// Issue all CHUNK rows of one 32 KB chunk with immediate offsets.
__device__ __forceinline__ void issue_chunk(unsigned la, unsigned long long ga) {
  issue_chunk_impl(la, ga, std::make_integer_sequence<int, CHUNK>{});
}

// Generic (partial-chunk) fallback: per-row computed addresses.
__device__ __forceinline__ void issue_partial(unsigned la, unsigned long long ga,
                                              int rows) {
  for (int m = 0; m < rows; ++m) {
    asm volatile("global_load_async_to_lds_b128 %0, %1, off th:TH_LOAD_NT"
                 :: "v"(la), "v"(ga) : "memory");
    la += ROWB;
    ga += ROWB;
  }
}

// Pass 1: per-node max timestamp (t_last) + occupancy mask.
// Timestamps are in [0,100): nonnegative, so uint bit-pattern order == float
// order, and zero-init equals the reference's sanitized value for empty nodes.
__global__ void tlast_mask_kernel(const float* __restrict__ ts,
                                  const int* __restrict__ ids,
                                  float* __restrict__ t_last,
                                  float* __restrict__ mask, int n) {
  int i = blockIdx.x * blockDim.x + threadIdx.x;
  if (i >= n) return;
  int s = ids[i];
  atomicMax((unsigned int*)(t_last + s), __float_as_uint(ts[i]));
  mask[s] = 1.0f;  // benign race: every writer stores 1.0f
}

// Pass 2: streaming weighted segment-sum.
// Block = 64 threads (2 waves). Wave w owns columns [w*128, w*128+128);
// each lane owns one float4. Messages staged to LDS via double-buffered async
// b128 copies (non-temporal: the 1 GB stream must not evict the L2-resident
// 102 MB output), weights broadcast with __shfl, register accumulator flushed
// with f32 atomics only at segment boundaries.
__global__ __launch_bounds__(64) void agg_kernel(
    const float* __restrict__ msgs, const float* __restrict__ ts,
    const int* __restrict__ ids, const float* __restrict__ t_last,
    float* __restrict__ agg, int total) {
  __shared__ float sbuf[2][CHUNK][D];  // 2 x 32 KB double buffer

  const int lane = threadIdx.x & 31;
  const int wv   = threadIdx.x >> 5;
  const int col  = wv * 128 + lane * 4;

  const int msg0 = blockIdx.x * SPAN;
  if (msg0 >= total) return;
  const int msg_end = min(msg0 + SPAN, total);
  const int nchunks = (msg_end - msg0 + CHUNK - 1) / CHUNK;

  // Loop-invariant LDS base addresses (this lane's 16 B within row 0).
  const unsigned la0 = (unsigned)(uintptr_t)(&sbuf[0][0][col]);
  const unsigned la1 = (unsigned)(uintptr_t)(&sbuf[1][0][col]);
  // Running global base address for the chunk currently being issued.
  unsigned long long ga =
      (unsigned long long)(uintptr_t)(msgs + (size_t)msg0 * D + col);

  // Prologue: issue chunk 0 into buffer 0.
  {
    const int rows0 = min(CHUNK, msg_end - msg0);
    if (rows0 == CHUNK) issue_chunk(la0, ga);
    else                issue_partial(la0, ga, rows0);
  }

  float4 acc = make_float4(0.f, 0.f, 0.f, 0.f);
  int cur_seg = -1;

  for (int c = 0; c < nchunks; ++c) {
    const int base = msg0 + c * CHUNK;
    const int rows = min(CHUNK, msg_end - base);
    const int buf  = c & 1;

    // Issue the next chunk into the other buffer before waiting (overlap).
    int rows_next = 0;
    if (c + 1 < nchunks) {
      ga += (unsigned long long)(CHUNK * ROWB);  // advance one chunk (32 KB)
      rows_next = min(CHUNK, msg_end - (base + CHUNK));
      const unsigned lan = buf ? la0 : la1;
      if (rows_next == CHUNK) issue_chunk(lan, ga);
      else                    issue_partial(lan, ga, rows_next);
    }
    // Async loads from one wave complete in order: ASYNCcnt <= rows_next
    // means everything issued for the *current* chunk has landed in LDS.
    if (rows_next == CHUNK) {
      asm volatile("s_wait_asynccnt 32" ::: "memory");
    } else {
      asm volatile("s_wait_asynccnt 0" ::: "memory");
    }

    // Per-message metadata: lane m computes (seg, weight) for message base+m.
    int myseg = 0;
    float myw = 0.f;
    if (lane < rows) {
      const int gi = base + lane;
      myseg = ids[gi];
      myw = expf((ts[gi] - t_last[myseg]) * 0.1f);
    }

    for (int m = 0; m < rows; ++m) {
      const int s   = __shfl(myseg, m, 32);  // wave-uniform
      const float w = __shfl(myw, m, 32);
      if (s != cur_seg) {                    // uniform branch within wave
        if (cur_seg >= 0) flush_acc(agg, cur_seg, col, acc);
        acc = make_float4(0.f, 0.f, 0.f, 0.f);
        cur_seg = s;
      }
      const float4 v = *(const float4*)(&sbuf[buf][m][col]);  // ds_load_b128
      acc.x = fmaf(w, v.x, acc.x);
      acc.y = fmaf(w, v.y, acc.y);
      acc.z = fmaf(w, v.z, acc.z);
      acc.w = fmaf(w, v.w, acc.w);
    }
  }
  if (cur_seg >= 0) flush_acc(agg, cur_seg, col, acc);
}

extern "C" void kernel_launch(void* const* d_in, const int* in_sizes, int n_in,
                              void* d_out, int out_size, void* d_ws, size_t ws_size,
                              hipStream_t stream) {
  const float* msgs = (const float*)d_in[0];
  const float* ts   = (const float*)d_in[1];
  const int*   ids  = (const int*)d_in[2];

  const int total     = in_sizes[1];          // number of messages
  const int num_nodes = out_size / (D + 2);   // agg + t_last + mask per node

  float* agg    = (float*)d_out;
  float* t_last = agg + (size_t)num_nodes * D;
  float* mask   = t_last + num_nodes;

  // Zero agg (atomic accumulation base), t_last (== sanitized value), mask.
  // Re-zeroing every call keeps kernel_launch deterministic across replays.
  hipMemsetAsync(d_out, 0, (size_t)out_size * sizeof(float), stream);

  const int b1 = (total + 255) / 256;
  tlast_mask_kernel<<<b1, 256, 0, stream>>>(ts, ids, t_last, mask, total);

  const int b2 = (total + SPAN - 1) / SPAN;
  agg_kernel<<<b2, 64, 0, stream>>>(msgs, ts, ids, t_last, agg, total);
}